// HubAlignLoss_74268574482705
// MI455X (gfx1250) — compile-verified
//
#include <hip/hip_runtime.h>

typedef __attribute__((ext_vector_type(2))) float v2f;
typedef __attribute__((ext_vector_type(8))) float v8f;

#define N_ROWS        262144
#define D_COLS        256
#define COLTILES      16                         // D / 16
#define ROWCHUNKS     128
#define ROWS_PER_CHUNK (N_ROWS / ROWCHUNKS)      // 2048
#define ITERS         (ROWS_PER_CHUNK / 4)       // 512 WMMAs per wave
#define WAVES_PER_BLOCK 8
#define NBLOCKS       ((COLTILES * ROWCHUNKS) / WAVES_PER_BLOCK) // 256

// ---------------------------------------------------------------------------
// Kernel 1: zero the 256-float column-sum accumulator in workspace
// ---------------------------------------------------------------------------
__global__ void hub_zero_ws(float* __restrict__ s) {
    s[threadIdx.x] = 0.0f;
}

// ---------------------------------------------------------------------------
// Kernel 2: column sums via ones(16x4) x B using V_WMMA_F32_16X16X4_F32.
// Each wave owns one 16-column tile and a 2048-row chunk. Per iteration it
// loads a 4x16 f32 slab of E into the B operand (lane L supplies column
// L&15 for rows {2h, 2h+1} of the slab, h = L>>4) and accumulates into C.
// With all-ones A, D[m][n] = sum_k B[k][n] regardless of K-slot permutation,
// and every row m of D is identical, so any C VGPR holds the column partials.
// ---------------------------------------------------------------------------
__global__ __launch_bounds__(256) void hub_colsum_wmma(
    const float* __restrict__ emb, float* __restrict__ s) {

    const int lane   = threadIdx.x & 31;
    const int wib    = threadIdx.x >> 5;
    const int w      = blockIdx.x * WAVES_PER_BLOCK + wib;
    const int coltile = w & (COLTILES - 1);
    const int chunk   = w >> 4;                 // log2(COLTILES)
    const int col0    = coltile * 16;
    const int n       = lane & 15;
    const int h       = lane >> 4;

    const size_t rowStart = (size_t)chunk * ROWS_PER_CHUNK;
    // Lane base: row (rowStart + 2h), column (col0 + n).
    const float* q = emb + (rowStart + (size_t)(2 * h)) * D_COLS + col0 + n;

    v2f a;  a.x = 1.0f; a.y = 1.0f;             // 16x4 all-ones A matrix
    v8f acc0 = {};                               // two accumulators to break
    v8f acc1 = {};                               // the WMMA RAW chain

    #pragma unroll 4
    for (int it = 0; it < ITERS; it += 2) {
        const float* p0 = q + (size_t)it * 4 * D_COLS;
        const float* p1 = p0 + 4 * D_COLS;
        v2f b0, b1;
        b0.x = p0[0];       // slab row 2h   (B VGPR0 slot for this lane)
        b0.y = p0[D_COLS];  // slab row 2h+1 (B VGPR1 slot)
        b1.x = p1[0];
        b1.y = p1[D_COLS];
        acc0 = __builtin_amdgcn_wmma_f32_16x16x4_f32(
                   false, a, false, b0, (short)0, acc0, false, false);
        acc1 = __builtin_amdgcn_wmma_f32_16x16x4_f32(
                   false, a, false, b1, (short)0, acc1, false, false);
    }

    // acc[0]: lanes 0-15 hold D[M=0][n], lanes 16-31 hold D[M=8][n-16];
    // all rows are equal (column sums), so take lane-half 0 only.
    float partial = acc0[0] + acc1[0];
    if (lane < 16) {
        atomicAdd(&s[col0 + n], partial);
    }
}

// ---------------------------------------------------------------------------
// Kernel 3: out = -kappa * p0 * dot(s, s)   (kappa == 1)
// ---------------------------------------------------------------------------
__global__ void hub_finalize(const float* __restrict__ s,
                             const float* __restrict__ p,
                             float* __restrict__ out) {
    __shared__ float red[256];
    const int t = threadIdx.x;
    const float v = s[t];
    red[t] = v * v;
    __syncthreads();
    #pragma unroll
    for (int off = 128; off > 0; off >>= 1) {
        if (t < off) red[t] += red[t + off];
        __syncthreads();
    }
    if (t == 0) out[0] = -p[0] * red[0];
}

// ---------------------------------------------------------------------------
extern "C" void kernel_launch(void* const* d_in, const int* in_sizes, int n_in,
                              void* d_out, int out_size, void* d_ws, size_t ws_size,
                              hipStream_t stream) {
    (void)in_sizes; (void)n_in; (void)out_size; (void)ws_size;
    const float* emb = (const float*)d_in[0];   // [262144, 256] f32
    const float* p   = (const float*)d_in[1];   // [1] f32
    float* out = (float*)d_out;                 // [1] f32
    float* s   = (float*)d_ws;                  // 256 f32 accumulator

    hipLaunchKernelGGL(hub_zero_ws,     dim3(1),       dim3(256), 0, stream, s);
    hipLaunchKernelGGL(hub_colsum_wmma, dim3(NBLOCKS), dim3(256), 0, stream, emb, s);
    hipLaunchKernelGGL(hub_finalize,    dim3(1),       dim3(256), 0, stream, s, p, out);
}